// CSATransformer_25778393710760
// MI455X (gfx1250) — compile-verified
//
#include <hip/hip_runtime.h>
#include <hip/hip_bf16.h>

typedef __attribute__((ext_vector_type(2))) float v2f;
typedef __attribute__((ext_vector_type(8))) float v8f;

#define BB 8
#define LL 4096
#define DXD 128

// ---------------------------------------------------------------------------
// K0: fold linear layers into rank-1 vectors.
//   u[d]   = sum_e W1[d,e]   * wt_w[e]
//   v1[d]  = sum_e Wsa1[d,e] * wsat_w[e]
//   v2[d]  = sum_e Wsa2[d,e] * wsat_w[e]
//   kq[b]  = sum_e ((c@W2)[b,e] + bsa[e]) * wt_w[e]
//   cc     = dot(bsa1, wsat_w) + wsat_b
// ---------------------------------------------------------------------------
__global__ __launch_bounds__(128) void k_prep(
    const float* __restrict__ W1, const float* __restrict__ W2,
    const float* __restrict__ wt_w, const float* __restrict__ bsa,
    const float* __restrict__ Wsa1, const float* __restrict__ Wsa2,
    const float* __restrict__ wsat_w, const float* __restrict__ wsat_b,
    const float* __restrict__ bsa1, const float* __restrict__ c,
    float* __restrict__ u, float* __restrict__ v1, float* __restrict__ v2,
    float* __restrict__ kq, float* __restrict__ cc)
{
    __shared__ float w2w[DXD];
    __shared__ float wt_s[DXD];
    __shared__ float wsat_s[DXD];
    const int t = threadIdx.x;
    wt_s[t] = wt_w[t];
    wsat_s[t] = wsat_w[t];
    __syncthreads();

    float su = 0.f, s1 = 0.f, s2 = 0.f, sw = 0.f;
    for (int e = 0; e < DXD; ++e) {
        su += W1[t * DXD + e] * wt_s[e];
        s1 += Wsa1[t * DXD + e] * wsat_s[e];
        s2 += Wsa2[t * DXD + e] * wsat_s[e];
        sw += W2[t * DXD + e] * wt_s[e];
    }
    u[t] = su; v1[t] = s1; v2[t] = s2; w2w[t] = sw;
    __syncthreads();

    if (t < BB) {
        float s = 0.f;
        for (int q = 0; q < DXD; ++q) s += c[t * DXD + q] * w2w[q];
        float bw = 0.f;
        for (int e = 0; e < DXD; ++e) bw += bsa[e] * wt_s[e];
        kq[t] = s + bw;
    }
    if (t == BB) {
        float s = wsat_b[0];
        for (int e = 0; e < DXD; ++e) s += bsa1[e] * wsat_s[e];
        cc[0] = s;
    }
}

// ---------------------------------------------------------------------------
// K1: per-row dot products (one wave32 per row).
//   logits[g] = x[g]·u + kq[b],  xv1[g] = x[g]·v1,  xv2[g] = x[g]·v2
// ---------------------------------------------------------------------------
__global__ __launch_bounds__(256) void k_dots(
    const float* __restrict__ x, const float* __restrict__ u,
    const float* __restrict__ v1, const float* __restrict__ v2,
    const float* __restrict__ kq,
    float* __restrict__ logits, float* __restrict__ xv1o, float* __restrict__ xv2o)
{
    const int wave = threadIdx.x >> 5;
    const int lane = threadIdx.x & 31;
    const long g = (long)blockIdx.x * 8 + wave;      // row in [0, B*L)
    const int b = (int)(g >> 12);                    // L = 4096

    const float4 xv = ((const float4*)(x + g * DXD))[lane];
    const float4 uu = ((const float4*)u)[lane];
    const float4 a1 = ((const float4*)v1)[lane];
    const float4 a2 = ((const float4*)v2)[lane];

    float du = xv.x * uu.x + xv.y * uu.y + xv.z * uu.z + xv.w * uu.w;
    float d1 = xv.x * a1.x + xv.y * a1.y + xv.z * a1.z + xv.w * a1.w;
    float d2 = xv.x * a2.x + xv.y * a2.y + xv.z * a2.z + xv.w * a2.w;
    #pragma unroll
    for (int o = 16; o > 0; o >>= 1) {
        du += __shfl_xor(du, o, 32);
        d1 += __shfl_xor(d1, o, 32);
        d2 += __shfl_xor(d2, o, 32);
    }
    if (lane == 0) {
        logits[g] = du + kq[b];
        xv1o[g] = d1;
        xv2o[g] = d2;
    }
}

// ---------------------------------------------------------------------------
// K2: softmax over L per batch; emit si = p*xv1, sj = p*xv2.
// ---------------------------------------------------------------------------
__global__ __launch_bounds__(512) void k_softmax(
    const float* __restrict__ logits, const float* __restrict__ xv1,
    const float* __restrict__ xv2,
    float* __restrict__ si, float* __restrict__ sj)
{
    __shared__ float red[512];
    const int tid = threadIdx.x;
    const long base = (long)blockIdx.x * LL;

    float vals[8];
    float lmax = -3.4e38f;
    #pragma unroll
    for (int i = 0; i < 8; ++i) {
        vals[i] = logits[base + tid + 512 * i];
        lmax = fmaxf(lmax, vals[i]);
    }
    red[tid] = lmax;
    __syncthreads();
    for (int s = 256; s > 0; s >>= 1) {
        if (tid < s) red[tid] = fmaxf(red[tid], red[tid + s]);
        __syncthreads();
    }
    lmax = red[0];
    __syncthreads();

    float e[8];
    float lsum = 0.f;
    #pragma unroll
    for (int i = 0; i < 8; ++i) {
        e[i] = __expf(vals[i] - lmax);
        lsum += e[i];
    }
    red[tid] = lsum;
    __syncthreads();
    for (int s = 256; s > 0; s >>= 1) {
        if (tid < s) red[tid] += red[tid + s];
        __syncthreads();
    }
    const float inv = 1.0f / red[0];

    #pragma unroll
    for (int i = 0; i < 8; ++i) {
        const long g = base + tid + 512 * i;
        const float p = e[i] * inv;
        si[g] = p * xv1[g];
        sj[g] = p * xv2[g];
    }
}

// ---------------------------------------------------------------------------
// K3: row denominators. D_i = sum_{j != i} exp(sigmoid(si_i + sj_j + C))
// One wave32 per row i; sj[b,:] staged in LDS. Stores 1/D_i.
// ---------------------------------------------------------------------------
__global__ __launch_bounds__(256) void k_rowsum(
    const float* __restrict__ si, const float* __restrict__ sj,
    const float* __restrict__ cc, float* __restrict__ dinv)
{
    __shared__ float sjl[LL];
    const int b = blockIdx.x >> 9;                 // 512 blocks per batch
    const int ibase = (blockIdx.x & 511) * 8;
    const long base = (long)b * LL;
    const float C = cc[0];

    for (int idx = threadIdx.x; idx < LL; idx += 256) sjl[idx] = sj[base + idx];
    __syncthreads();

    const int wave = threadIdx.x >> 5;
    const int lane = threadIdx.x & 31;
    const int i = ibase + wave;
    const float sii = si[base + i];

    float acc = 0.f;
    for (int j = lane; j < LL; j += 32) {
        const float z = sii + sjl[j] + C;
        const float sg = 1.0f / (1.0f + __expf(-z));
        acc += __expf(sg);
    }
    #pragma unroll
    for (int o = 16; o > 0; o >>= 1) acc += __shfl_xor(acc, o, 32);
    if (lane == 0) {
        const float zd = sii + sjl[i] + C;
        const float sgd = 1.0f / (1.0f + __expf(-zd));
        dinv[base + i] = 1.0f / (acc - __expf(sgd));
    }
}

// ---------------------------------------------------------------------------
// K4: column sums of A. colsum_j = sum_{i != j} exp(sigmoid(si_i+sj_j+C))/D_i
// One wave32 per column j; si[b,:] and 1/D[b,:] staged in LDS.
// ---------------------------------------------------------------------------
__global__ __launch_bounds__(256) void k_colsum(
    const float* __restrict__ si, const float* __restrict__ sj,
    const float* __restrict__ dinv, const float* __restrict__ cc,
    float* __restrict__ colsum)
{
    __shared__ float sil[LL];
    __shared__ float dl[LL];
    const int b = blockIdx.x >> 9;
    const int jbase = (blockIdx.x & 511) * 8;
    const long base = (long)b * LL;
    const float C = cc[0];

    for (int idx = threadIdx.x; idx < LL; idx += 256) {
        sil[idx] = si[base + idx];
        dl[idx] = dinv[base + idx];
    }
    __syncthreads();

    const int wave = threadIdx.x >> 5;
    const int lane = threadIdx.x & 31;
    const int j = jbase + wave;
    const float sjj = sj[base + j];

    float acc = 0.f;
    for (int i = lane; i < LL; i += 32) {
        const float z = sil[i] + sjj + C;
        const float sg = 1.0f / (1.0f + __expf(-z));
        acc += __expf(sg) * dl[i];
    }
    #pragma unroll
    for (int o = 16; o > 0; o >>= 1) acc += __shfl_xor(acc, o, 32);
    if (lane == 0) {
        const float zd = sil[j] + sjj + C;
        const float sgd = 1.0f / (1.0f + __expf(-zd));
        colsum[base + j] = acc - __expf(sgd) * dl[j];
    }
}

// ---------------------------------------------------------------------------
// K5: fused FFN + residual + LayerNorm using V_WMMA_F32_16X16X4_F32.
//   ui = x * colsum ;  y = relu(ui@W1f + b1)@W2f + b2 + ui ;  LN(y)
// Block = 256 threads (8 waves), 64 rows/block. Each wave owns a 16-wide
// N-slice and loops 4 M-tiles (B operand reused 4x per WMMA k-step).
// LDS row pitch 132 floats: 16B aligned + bank-conflict-free WMMA A loads.
// ---------------------------------------------------------------------------
#define PITCH 132

__global__ __launch_bounds__(256) void k_ffn_ln(
    const float* __restrict__ x, const float* __restrict__ colsum,
    const float* __restrict__ w1, const float* __restrict__ b1,
    const float* __restrict__ w2, const float* __restrict__ b2,
    const float* __restrict__ ln_g, const float* __restrict__ ln_b,
    float* __restrict__ out)
{
    __shared__ float Hs[64 * PITCH];
    __shared__ float Ys[64 * PITCH];
    __shared__ float mus[64];
    __shared__ float rstds[64];

    const int tid = threadIdx.x;
    const long rowbase = (long)blockIdx.x * 64;

    // Stage H = x * colsum (coalesced float4 loads).
    for (int idx = tid; idx < 64 * 32; idx += 256) {
        const int m = idx >> 5;
        const int q = idx & 31;
        const float4 xv = ((const float4*)(x + (rowbase + m) * DXD))[q];
        const float s = colsum[rowbase + m];
        float4 hv;
        hv.x = xv.x * s; hv.y = xv.y * s; hv.z = xv.z * s; hv.w = xv.w * s;
        *(float4*)&Hs[m * PITCH + q * 4] = hv;
    }
    __syncthreads();

    const int lane = tid & 31;
    const int wave = tid >> 5;
    const int mrow = lane & 15;          // A operand row (lanes 0-15 == 16-31)
    const int khalf = (lane >> 4) * 2;   // K sub-offset 0 or 2
    const int ncol = wave * 16 + (lane & 15);

    // ---- GEMM1: Y1 = relu(H @ W1 + b1) ----
    v8f acc0 = 0.0f, acc1 = 0.0f, acc2 = 0.0f, acc3 = 0.0f;
    for (int s = 0; s < 32; ++s) {
        const int k = 4 * s + khalf;
        v2f bv;
        bv.x = w1[k * DXD + ncol];
        bv.y = w1[(k + 1) * DXD + ncol];
        const v2f a0 = *(const v2f*)&Hs[(mrow +  0) * PITCH + k];
        const v2f a1 = *(const v2f*)&Hs[(mrow + 16) * PITCH + k];
        const v2f a2 = *(const v2f*)&Hs[(mrow + 32) * PITCH + k];
        const v2f a3 = *(const v2f*)&Hs[(mrow + 48) * PITCH + k];
        acc0 = __builtin_amdgcn_wmma_f32_16x16x4_f32(false, a0, false, bv, (short)0, acc0, false, false);
        acc1 = __builtin_amdgcn_wmma_f32_16x16x4_f32(false, a1, false, bv, (short)0, acc1, false, false);
        acc2 = __builtin_amdgcn_wmma_f32_16x16x4_f32(false, a2, false, bv, (short)0, acc2, false, false);
        acc3 = __builtin_amdgcn_wmma_f32_16x16x4_f32(false, a3, false, bv, (short)0, acc3, false, false);
    }
    {
        const float bias1 = b1[ncol];
        const int mloc = 8 * (lane >> 4);    // C/D layout: row = r + 8*(lane/16)
        #pragma unroll
        for (int r = 0; r < 8; ++r) {
            const int m = mloc + r;
            Ys[(m +  0) * PITCH + ncol] = fmaxf(acc0[r] + bias1, 0.0f);
            Ys[(m + 16) * PITCH + ncol] = fmaxf(acc1[r] + bias1, 0.0f);
            Ys[(m + 32) * PITCH + ncol] = fmaxf(acc2[r] + bias1, 0.0f);
            Ys[(m + 48) * PITCH + ncol] = fmaxf(acc3[r] + bias1, 0.0f);
        }
    }
    __syncthreads();

    // ---- GEMM2: Y = Y1 @ W2 ----
    acc0 = 0.0f; acc1 = 0.0f; acc2 = 0.0f; acc3 = 0.0f;
    for (int s = 0; s < 32; ++s) {
        const int k = 4 * s + khalf;
        v2f bv;
        bv.x = w2[k * DXD + ncol];
        bv.y = w2[(k + 1) * DXD + ncol];
        const v2f a0 = *(const v2f*)&Ys[(mrow +  0) * PITCH + k];
        const v2f a1 = *(const v2f*)&Ys[(mrow + 16) * PITCH + k];
        const v2f a2 = *(const v2f*)&Ys[(mrow + 32) * PITCH + k];
        const v2f a3 = *(const v2f*)&Ys[(mrow + 48) * PITCH + k];
        acc0 = __builtin_amdgcn_wmma_f32_16x16x4_f32(false, a0, false, bv, (short)0, acc0, false, false);
        acc1 = __builtin_amdgcn_wmma_f32_16x16x4_f32(false, a1, false, bv, (short)0, acc1, false, false);
        acc2 = __builtin_amdgcn_wmma_f32_16x16x4_f32(false, a2, false, bv, (short)0, acc2, false, false);
        acc3 = __builtin_amdgcn_wmma_f32_16x16x4_f32(false, a3, false, bv, (short)0, acc3, false, false);
    }
    __syncthreads();   // all waves finished reading Ys -> safe to overwrite

    // Epilogue: + b2 + residual(H), store Y into Ys.
    {
        const float bias2 = b2[ncol];
        const int mloc = 8 * (lane >> 4);
        #pragma unroll
        for (int r = 0; r < 8; ++r) {
            const int m = mloc + r;
            Ys[(m +  0) * PITCH + ncol] = acc0[r] + bias2 + Hs[(m +  0) * PITCH + ncol];
            Ys[(m + 16) * PITCH + ncol] = acc1[r] + bias2 + Hs[(m + 16) * PITCH + ncol];
            Ys[(m + 32) * PITCH + ncol] = acc2[r] + bias2 + Hs[(m + 32) * PITCH + ncol];
            Ys[(m + 48) * PITCH + ncol] = acc3[r] + bias2 + Hs[(m + 48) * PITCH + ncol];
        }
    }
    __syncthreads();

    // LayerNorm stats (one thread per row).
    if (tid < 64) {
        float sum = 0.f, sumsq = 0.f;
        for (int d = 0; d < DXD; ++d) {
            const float v = Ys[tid * PITCH + d];
            sum += v;
            sumsq += v * v;
        }
        const float mu = sum * (1.0f / DXD);
        const float var = sumsq * (1.0f / DXD) - mu * mu;
        mus[tid] = mu;
        rstds[tid] = rsqrtf(var + 1e-6f);
    }
    __syncthreads();

    // Coalesced normalize + affine + store.
    for (int idx = tid; idx < 64 * DXD; idx += 256) {
        const int m = idx >> 7;
        const int d = idx & 127;
        const float v = (Ys[m * PITCH + d] - mus[m]) * rstds[m];
        out[(rowbase + m) * DXD + d] = v * ln_g[d] + ln_b[d];
    }
}

// ---------------------------------------------------------------------------
extern "C" void kernel_launch(void* const* d_in, const int* in_sizes, int n_in,
                              void* d_out, int out_size, void* d_ws, size_t ws_size,
                              hipStream_t stream) {
    const float* x      = (const float*)d_in[0];
    const float* c      = (const float*)d_in[1];
    const float* W1     = (const float*)d_in[2];
    const float* W2     = (const float*)d_in[3];
    const float* wt_w   = (const float*)d_in[4];
    const float* bsa    = (const float*)d_in[5];
    const float* Wsa1   = (const float*)d_in[6];
    const float* Wsa2   = (const float*)d_in[7];
    const float* wsat_w = (const float*)d_in[8];
    const float* wsat_b = (const float*)d_in[9];
    const float* bsa1   = (const float*)d_in[10];
    const float* pfn_w1 = (const float*)d_in[11];
    const float* pfn_b1 = (const float*)d_in[12];
    const float* pfn_w2 = (const float*)d_in[13];
    const float* pfn_b2 = (const float*)d_in[14];
    const float* ln_g   = (const float*)d_in[15];
    const float* ln_b   = (const float*)d_in[16];

    float* ws = (float*)d_ws;
    const int N = BB * LL;                 // 32768 rows
    float* u      = ws;                    // 128
    float* v1     = ws + 128;              // 128
    float* v2     = ws + 256;              // 128
    float* kq     = ws + 384;              // 8
    float* cc     = ws + 392;              // 1
    float* logits = ws + 512;              // N
    float* xv1    = logits + N;            // N
    float* xv2    = xv1 + N;               // N
    float* si     = xv2 + N;               // N
    float* sj     = si + N;                // N
    float* dinv   = sj + N;                // N
    float* colsum = dinv + N;              // N

    k_prep<<<1, 128, 0, stream>>>(W1, W2, wt_w, bsa, Wsa1, Wsa2, wsat_w, wsat_b,
                                  bsa1, c, u, v1, v2, kq, cc);
    k_dots<<<N / 8, 256, 0, stream>>>(x, u, v1, v2, kq, logits, xv1, xv2);
    k_softmax<<<BB, 512, 0, stream>>>(logits, xv1, xv2, si, sj);
    k_rowsum<<<N / 8, 256, 0, stream>>>(si, sj, cc, dinv);
    k_colsum<<<N / 8, 256, 0, stream>>>(si, sj, dinv, cc, colsum);
    k_ffn_ln<<<N / 64, 256, 0, stream>>>(x, colsum, pfn_w1, pfn_b1, pfn_w2, pfn_b2,
                                         ln_g, ln_b, (float*)d_out);
}